// PathfinderDiscoveryNetwork_52819507806473
// MI455X (gfx1250) — compile-verified
//
#include <hip/hip_runtime.h>
#include <math.h>

typedef __attribute__((ext_vector_type(2))) float v2f;
typedef __attribute__((ext_vector_type(8))) float v8f;

// ---------------------------------------------------------------------------
// Generic fill
// ---------------------------------------------------------------------------
__global__ void fill_kernel(float* __restrict__ p, float v, int n) {
    int i = blockIdx.x * blockDim.x + threadIdx.x;
    if (i < n) p[i] = v;
}

// ---------------------------------------------------------------------------
// Edge MLP: ew[e] = sigmoid( relu(edge_x[e] @ W1 + b1) @ W2 + b2 )
// edge_x: [E,16], W1: [16,32], b1: [32], W2: [32,1], b2: [1]
// ---------------------------------------------------------------------------
__global__ __launch_bounds__(256) void edge_mlp_kernel(
    const float* __restrict__ ex, const float* __restrict__ W1,
    const float* __restrict__ b1, const float* __restrict__ W2,
    const float* __restrict__ b2, float* __restrict__ ew, int E) {
    __shared__ float sW1[16 * 32];
    __shared__ float sb1[32];
    __shared__ float sW2[32];
    __shared__ float sb2;
    for (int i = threadIdx.x; i < 512; i += 256) sW1[i] = W1[i];
    if (threadIdx.x < 32) { sb1[threadIdx.x] = b1[threadIdx.x]; sW2[threadIdx.x] = W2[threadIdx.x]; }
    if (threadIdx.x == 0) sb2 = b2[0];
    __syncthreads();

    int e = blockIdx.x * 256 + threadIdx.x;
    if (e >= E) return;
    float xin[16];
    const float* p = ex + (size_t)e * 16;
#pragma unroll
    for (int k = 0; k < 16; ++k) xin[k] = p[k];
    float s = sb2;
#pragma unroll
    for (int j = 0; j < 32; ++j) {
        float acc = sb1[j];
#pragma unroll
        for (int k = 0; k < 16; ++k) acc = fmaf(xin[k], sW1[k * 32 + j], acc);
        s = fmaf(fmaxf(acc, 0.0f), sW2[j], s);
    }
    ew[e] = 1.0f / (1.0f + expf(-s));
}

// ---------------------------------------------------------------------------
// Degree accumulation: deg[col[e]] += ew[e]   (deg pre-initialized to 1.0 for
// the self-loop weight)
// ---------------------------------------------------------------------------
__global__ void deg_kernel(const long long* __restrict__ col,
                           const float* __restrict__ ew,
                           float* __restrict__ deg, int E) {
    int e = blockIdx.x * blockDim.x + threadIdx.x;
    if (e < E) atomicAdd(&deg[(int)col[e]], ew[e]);
}

__global__ void dinv_kernel(const float* __restrict__ deg,
                            float* __restrict__ dinv, int n) {
    int i = blockIdx.x * blockDim.x + threadIdx.x;
    if (i < n) {
        float d = deg[i];
        dinv[i] = (d > 0.0f) ? rsqrtf(d) : 0.0f;
    }
}

// ---------------------------------------------------------------------------
// f32 GEMM via V_WMMA_F32_16X16X4_F32.  C[M,NCOL] = A[M,K] @ B[K,NCOL]
// One wave computes one 16x16 output tile, stepping K by 4.
// A-frag (16x4):  lane m=lane&15, holds K = 2*(lane>>4)+{0,1} in v[0],v[1]
// B-frag (4x16):  vgpr v, lane n=lane&15, holds K = v + 2*(lane>>4)
// C/D (16x16):    vgpr r, lanes0-15 -> M=r, lanes16-31 -> M=r+8, N=lane&15
// ---------------------------------------------------------------------------
template <int K, int NCOL>
__global__ __launch_bounds__(256) void gemm_wmma_f32(
    const float* __restrict__ A, const float* __restrict__ B,
    float* __restrict__ C, int M) {
    constexpr int NT = (NCOL + 15) / 16;
    int gw   = (int)((blockIdx.x * (unsigned)blockDim.x + threadIdx.x) >> 5);
    int lane = threadIdx.x & 31;
    int mt = gw / NT;
    int nt = gw % NT;
    if (mt * 16 >= M) return;  // wave-uniform exit

    int lo = lane & 15;
    int hi = lane >> 4;
    int m = mt * 16 + lo;
    int n = nt * 16 + lo;
    bool nok = (NCOL % 16 == 0) || (n < NCOL);

    v8f acc = {0.f, 0.f, 0.f, 0.f, 0.f, 0.f, 0.f, 0.f};
#pragma unroll 4
    for (int k0 = 0; k0 < K; k0 += 4) {
        int ka = k0 + 2 * hi;
        v2f a = *(const v2f*)(A + (size_t)m * K + ka);
        v2f b;
        b[0] = nok ? B[(size_t)(ka + 0) * NCOL + n] : 0.0f;
        b[1] = nok ? B[(size_t)(ka + 1) * NCOL + n] : 0.0f;
        acc = __builtin_amdgcn_wmma_f32_16x16x4_f32(
            /*neg_a=*/false, a, /*neg_b=*/false, b,
            /*c_mod=*/(short)0, acc, /*reuse_a=*/false, /*reuse_b=*/false);
    }
#pragma unroll
    for (int r = 0; r < 8; ++r) {
        int mm = mt * 16 + r + 8 * hi;
        if (nok) C[(size_t)mm * NCOL + n] = acc[r];
    }
}

// ---------------------------------------------------------------------------
// Edge scatter: out[col[e]] += dinv[row]*ew[e]*dinv[col] * h[row[e]]
// One wave per edge. F=128: lane handles a float4. F=40: lane handles
// features lane and lane+32 (masked).
// ---------------------------------------------------------------------------
template <int F>
__global__ __launch_bounds__(256) void scatter_kernel(
    const long long* __restrict__ row, const long long* __restrict__ col,
    const float* __restrict__ ew, const float* __restrict__ dinv,
    const float* __restrict__ h, float* __restrict__ out, int E) {
    int wid  = (int)((blockIdx.x * (unsigned)blockDim.x + threadIdx.x) >> 5);
    int lane = threadIdx.x & 31;
    if (wid >= E) return;
    int r = (int)row[wid];
    int c = (int)col[wid];
    float nrm = dinv[r] * ew[wid] * dinv[c];
    if (F == 128) {
        const float4 v = *(const float4*)(h + (size_t)r * F + lane * 4);
        float* o = out + (size_t)c * F + lane * 4;
        atomicAdd(o + 0, v.x * nrm);
        atomicAdd(o + 1, v.y * nrm);
        atomicAdd(o + 2, v.z * nrm);
        atomicAdd(o + 3, v.w * nrm);
    } else {
        if (lane < F)
            atomicAdd(out + (size_t)c * F + lane, h[(size_t)r * F + lane] * nrm);
        if (lane + 32 < F)
            atomicAdd(out + (size_t)c * F + lane + 32,
                      h[(size_t)r * F + lane + 32] * nrm);
    }
}

// ---------------------------------------------------------------------------
// Finalize conv1: agg = relu(agg + dinv[i]^2 * h1 + bg1)   (self loop + bias)
// ---------------------------------------------------------------------------
__global__ void finalize1_kernel(const float* __restrict__ h1,
                                 const float* __restrict__ dinv,
                                 const float* __restrict__ bg1,
                                 float* __restrict__ agg, int total) {
    int i = blockIdx.x * blockDim.x + threadIdx.x;
    if (i >= total) return;
    int node = i >> 7;        // /128
    int c    = i & 127;
    float di = dinv[node];
    float v  = agg[i] + di * di * h1[i] + bg1[c];
    agg[i] = fmaxf(v, 0.0f);
}

// Finalize conv2: out += dinv[i]^2 * h2 + bg2  (no relu on final layer)
__global__ void finalize2_kernel(const float* __restrict__ h2,
                                 const float* __restrict__ dinv,
                                 const float* __restrict__ bg2,
                                 float* __restrict__ out, int total, int cls) {
    int i = blockIdx.x * blockDim.x + threadIdx.x;
    if (i >= total) return;
    int node = i / cls;
    int c    = i % cls;
    float di = dinv[node];
    out[i] = out[i] + di * di * h2[i] + bg2[c];
}

// ---------------------------------------------------------------------------
// Launch
// ---------------------------------------------------------------------------
extern "C" void kernel_launch(void* const* d_in, const int* in_sizes, int n_in,
                              void* d_out, int out_size, void* d_ws, size_t ws_size,
                              hipStream_t stream) {
    const float*     x    = (const float*)d_in[0];
    const long long* eidx = (const long long*)d_in[1];   // int64 [2,E]
    const float*     ex   = (const float*)d_in[2];
    const float*     W1   = (const float*)d_in[3];
    const float*     b1   = (const float*)d_in[4];
    const float*     W2   = (const float*)d_in[5];
    const float*     b2   = (const float*)d_in[6];
    const float*     Wg1  = (const float*)d_in[7];
    const float*     bg1  = (const float*)d_in[8];
    const float*     Wg2  = (const float*)d_in[9];
    const float*     bg2  = (const float*)d_in[10];
    float* out = (float*)d_out;

    const int NF  = 128;
    const int CLS = 40;
    const int EF  = 16;
    const int Nn  = in_sizes[0] / NF;   // 50000
    const int E   = in_sizes[2] / EF;   // 1.6M

    const long long* row = eidx;        // edge_index[0]
    const long long* col = eidx + E;    // edge_index[1]

    // Workspace carve-up (float units, 64B-aligned segments)
    float* ws = (float*)d_ws;
    size_t off = 0;
    auto carve = [&](size_t nfl) {
        float* p = ws + off;
        off += (nfl + 15) & ~(size_t)15;
        return p;
    };
    float* ew   = carve((size_t)E);
    float* deg  = carve((size_t)Nn);
    float* dinv = carve((size_t)Nn);
    float* h1   = carve((size_t)Nn * NF);
    float* agg1 = carve((size_t)Nn * NF);
    float* h2   = carve((size_t)Nn * CLS);
    (void)ws_size;

    auto cdiv = [](long long a, long long b) { return (int)((a + b - 1) / b); };
    const int TB = 256;

    // 0) initialize accumulators
    fill_kernel<<<cdiv(Nn, TB), TB, 0, stream>>>(deg, 1.0f, Nn);                 // self loop wt
    fill_kernel<<<cdiv((long long)Nn * NF, TB), TB, 0, stream>>>(agg1, 0.0f, Nn * NF);
    fill_kernel<<<cdiv(out_size, TB), TB, 0, stream>>>(out, 0.0f, out_size);

    // 1) edge-weight MLP
    edge_mlp_kernel<<<cdiv(E, TB), TB, 0, stream>>>(ex, W1, b1, W2, b2, ew, E);

    // 2) degrees + rsqrt
    deg_kernel<<<cdiv(E, TB), TB, 0, stream>>>(col, ew, deg, E);
    dinv_kernel<<<cdiv(Nn, TB), TB, 0, stream>>>(deg, dinv, Nn);

    // 3) h1 = x @ Wg1  (WMMA f32)
    {
        int mt = cdiv(Nn, 16);
        int waves = mt * (NF / 16);
        gemm_wmma_f32<128, 128><<<cdiv((long long)waves * 32, TB), TB, 0, stream>>>(
            x, Wg1, h1, Nn);
    }

    // 4) conv1 scatter, then self-loop + bias + relu
    scatter_kernel<128><<<cdiv((long long)E * 32, TB), TB, 0, stream>>>(
        row, col, ew, dinv, h1, agg1, E);
    finalize1_kernel<<<cdiv((long long)Nn * NF, TB), TB, 0, stream>>>(
        h1, dinv, bg1, agg1, Nn * NF);

    // 5) h2 = relu(h) @ Wg2  (WMMA f32, N=40 padded to 48)
    {
        int mt = cdiv(Nn, 16);
        int waves = mt * ((CLS + 15) / 16);
        gemm_wmma_f32<128, 40><<<cdiv((long long)waves * 32, TB), TB, 0, stream>>>(
            agg1, Wg2, h2, Nn);
    }

    // 6) conv2 scatter into d_out, then self-loop + bias
    scatter_kernel<40><<<cdiv((long long)E * 32, TB), TB, 0, stream>>>(
        row, col, ew, dinv, h2, out, E);
    finalize2_kernel<<<cdiv((long long)Nn * CLS, TB), TB, 0, stream>>>(
        h2, dinv, bg2, out, Nn * CLS, CLS);
}